// AFRM_27573690040558
// MI455X (gfx1250) — compile-verified
//
#include <hip/hip_runtime.h>
#include <math.h>

typedef float v2f __attribute__((ext_vector_type(2)));
typedef float v8f __attribute__((ext_vector_type(8)));

#define CC 64
#define KS_ 7
#define AA 16
#define BB 4
#define HH_ 128
#define WW 128
#define HWS (HH_*WW)
#define EPSF 1e-5f

__device__ __forceinline__ float sigmf(float x){ return 1.0f/(1.0f+expf(-x)); }
__device__ __forceinline__ int mini(int a,int b){ return a<b?a:b; }
__device__ __forceinline__ int maxi(int a,int b){ return a>b?a:b; }

// ---------------- K1: global average pool g[b][c] ----------------
__global__ void kpool(const float* __restrict__ x, float* __restrict__ g){
  __shared__ float red[256];
  int bc = blockIdx.x, t = threadIdx.x;
  const float* p = x + (size_t)bc*HWS;
  float s = 0.f;
  for(int i=t;i<HWS;i+=256) s += p[i];
  red[t]=s; __syncthreads();
  for(int o=128;o>0;o>>=1){ if(t<o) red[t]+=red[t+o]; __syncthreads(); }
  if(t==0) g[bc] = red[0]*(1.0f/HWS);
}

// ---------------- K2: attention vectors (both omni blocks) + zero GN sums ----------------
__global__ void katt(const float* __restrict__ g,
  const float* fc1, const float* g1, const float* b1, const float* m1, const float* v1,
  const float* chw1, const float* chb1, const float* flw1, const float* flb1,
  const float* fc2, const float* g2, const float* b2, const float* m2, const float* v2,
  const float* chw2, const float* chb2, const float* flw2, const float* flb2,
  const float* spw2, const float* spb2,
  float* c1o, float* f1o, float* c2o, float* f2o, float* sp2o, float* gnbuf){
  __shared__ float a1[BB*AA], a2[BB*AA];
  int t = threadIdx.x;
  if(t < BB*AA){
    int b = t>>4, i = t&15;
    float d=0.f; for(int c=0;c<CC;c++) d += g[b*CC+c]*fc1[i*CC+c];
    d = (d - m1[i])*rsqrtf(v1[i]+EPSF)*g1[i] + b1[i];
    a1[t] = fmaxf(d, 0.f);
  } else if(t < 2*BB*AA){
    int tt = t - BB*AA; int b = tt>>4, i = tt&15;
    float d=0.f; for(int c=0;c<CC;c++) d += g[b*CC+c]*fc2[i*CC+c];
    d = (d - m2[i])*rsqrtf(v2[i]+EPSF)*g2[i] + b2[i];
    a2[tt] = fmaxf(d, 0.f);
  }
  __syncthreads();
  {
    int b = t>>6, co = t&63;
    float d1=chb1[co], d2=flb1[co], d3=chb2[co], d4=flb2[co];
    for(int i=0;i<AA;i++){
      float x1 = a1[b*AA+i], x2 = a2[b*AA+i];
      d1 += x1*chw1[co*AA+i];
      d2 += x1*flw1[co*AA+i];
      d3 += x2*chw2[co*AA+i];
      d4 += x2*flw2[co*AA+i];
    }
    c1o[t]=sigmf(d1); f1o[t]=sigmf(d2); c2o[t]=sigmf(d3); f2o[t]=sigmf(d4);
  }
  if(t < BB*9){
    int b=t/9, j=t%9;
    float d=spb2[j];
    for(int i=0;i<AA;i++) d += a2[b*AA+i]*spw2[j*AA+i];
    sp2o[b*16+j]=sigmf(d);
  }
  if(t < 128) gnbuf[t]=0.f;   // zeroes gnsum[64] + gnsq[64]
}

// ---------------- K3: effective dynamic kernels weff[b][o][3][3] ----------------
__global__ void kweff(const float* __restrict__ wgt, const float* __restrict__ ratio_p,
                      const float* __restrict__ c1, const float* __restrict__ f1,
                      const float* __restrict__ c2, const float* __restrict__ f2,
                      const float* __restrict__ sp2, float* __restrict__ weff){
  __shared__ float c1s[BB*CC], c2s[BB*CC];
  int t = threadIdx.x;
  c1s[t]=c1[t]; c2s[t]=c2[t];
  __syncthreads();
  int b = t>>6, o = t&63;
  float ratio = ratio_p[0];
  float s1 = 0.f; float s2[9];
  #pragma unroll
  for(int e=0;e<9;e++) s2[e]=0.f;
  for(int j=0;j<CC;j++){
    const float* wp = wgt + ((size_t)(o*CC+j))*9;
    float wv9[9]; float wsum=0.f;
    #pragma unroll
    for(int e=0;e<9;e++){ wv9[e]=wp[e]; wsum+=wv9[e]; }
    float wm = wsum*(1.f/9.f);
    s1 += wm*c1s[b*CC+j];
    float cj = c2s[b*CC+j];
    #pragma unroll
    for(int e=0;e<9;e++) s2[e] += (wv9[e]-wm)*cj;
  }
  const float Dm[9] = {2.f,2.f,2.f, 1.73205081f,0.f,-1.73205081f, 1.f,-2.f,1.f};
  const float Di[9] = {0.16666667f,0.28867513f,0.16666667f,
                       0.16666667f,0.f,-0.33333333f,
                       0.16666667f,-0.28867513f,0.16666667f};
  float t1[9], dct[9], t2[9];
  for(int k=0;k<3;k++) for(int n=0;n<3;n++){
    float a=0.f; for(int m=0;m<3;m++) a += Dm[k*3+m]*s2[m*3+n];
    t1[k*3+n]=a;
  }
  for(int k=0;k<3;k++) for(int l=0;l<3;l++){
    float a=0.f; for(int n=0;n<3;n++) a += t1[k*3+n]*Dm[l*3+n];
    dct[k*3+l] = a * 2.f * sp2[b*16 + k*3+l];
  }
  for(int k=0;k<3;k++) for(int n=0;n<3;n++){
    float a=0.f; for(int m=0;m<3;m++) a += Di[k*3+m]*dct[m*3+n];
    t2[k*3+n]=a;
  }
  float base = ratio*4.f*f1[t]*s1;
  float sc   = (1.f-ratio)*4.f*f2[t];
  for(int k=0;k<3;k++) for(int l=0;l<3;l++){
    float a=0.f; for(int n=0;n<3;n++) a += t2[k*3+n]*Di[l*3+n];
    weff[(size_t)t*9 + k*3+l] = base + sc*a;
  }
}

// ---------------- K4: HH depthwise conv + IWT + residual -> out ----------------
__global__ void kout(const float* __restrict__ x, const float* __restrict__ weff,
                     float* __restrict__ outb){
  __shared__ float hhl[64*65];
  int bc = blockIdx.x, t = threadIdx.x;
  const float* xc = x + (size_t)bc*HWS;
  float wf[9];
  #pragma unroll
  for(int e=0;e<9;e++) wf[e]=weff[(size_t)bc*9+e];
  for(int r=0;r<16;r++){
    int idx=t+r*256; int i=idx>>6, j=idx&63;
    float p1=xc[(2*i)*WW+2*j],   p2=xc[(2*i+1)*WW+2*j];
    float p3=xc[(2*i)*WW+2*j+1], p4=xc[(2*i+1)*WW+2*j+1];
    hhl[i*65+j]=0.5f*(p1-p2-p3+p4);
  }
  __syncthreads();
  float* oc = outb + (size_t)bc*HWS;
  for(int r=0;r<16;r++){
    int idx=t+r*256; int i=idx>>6, j=idx&63;
    float acc=0.f;
    #pragma unroll
    for(int di=0;di<3;di++){
      int ii=i+di-1; if(ii<0||ii>63) continue;
      #pragma unroll
      for(int dj=0;dj<3;dj++){
        int jj=j+dj-1; if(jj<0||jj>63) continue;
        acc += wf[di*3+dj]*hhl[ii*65+jj];
      }
    }
    float p1=xc[(2*i)*WW+2*j],   p2=xc[(2*i+1)*WW+2*j];
    float p3=xc[(2*i)*WW+2*j+1], p4=xc[(2*i+1)*WW+2*j+1];
    float LL=0.5f*(p1+p2+p3+p4), LH=0.5f*(-p1+p2-p3+p4), HL=0.5f*(-p1-p2+p3+p4);
    float a=LL*0.5f, bb=LH*0.5f, cc=HL*0.5f, dd=acc*0.5f;
    oc[(2*i)*WW+2*j]     = a-bb-cc+dd + p1;
    oc[(2*i+1)*WW+2*j]   = a+bb-cc-dd + p2;
    oc[(2*i)*WW+2*j+1]   = a-bb+cc-dd + p3;
    oc[(2*i+1)*WW+2*j+1] = a+bb+cc+dd + p4;
  }
}

// ---------------- K5: offset conv (7 ch) + BN + tanh + cumsum -> dy ----------------
__global__ void koff(const float* __restrict__ outb, const float* __restrict__ offw,
                     const float* __restrict__ offb, const float* __restrict__ bng,
                     const float* __restrict__ bnb, const float* __restrict__ bnm,
                     const float* __restrict__ bnv, float* __restrict__ dy){
  __shared__ float wl[7*CC*9];
  __shared__ float til[18*20];
  int b = blockIdx.z, h0 = blockIdx.y*16, w0 = blockIdx.x*16;
  int t = threadIdx.x;
  for(int e=t;e<7*CC*9;e+=256) wl[e]=offw[e];   // first 7 output channels, contiguous
  float acc[7];
  #pragma unroll
  for(int k=0;k<7;k++) acc[k]=0.f;
  int lr = t>>4, lc = t&15;
  for(int ci=0;ci<CC;ci++){
    __syncthreads();
    const float* ic = outb + ((size_t)(b*CC+ci))*HWS;
    for(int e=t;e<324;e+=256){
      int r=e/18, c=e-r*18;
      int gh=h0-1+r, gw=w0-1+c;
      til[r*20+c] = (gh>=0&&gh<HH_&&gw>=0&&gw<WW) ? ic[gh*WW+gw] : 0.f;
    }
    __syncthreads();
    #pragma unroll
    for(int kh=0;kh<3;kh++)
      #pragma unroll
      for(int kw=0;kw<3;kw++){
        float v = til[(lr+kh)*20 + lc+kw];
        const float* wp = wl + ci*9 + kh*3+kw;
        #pragma unroll
        for(int oc_=0;oc_<7;oc_++) acc[oc_] += v*wp[oc_*CC*9];
      }
  }
  float y[7];
  #pragma unroll
  for(int k=0;k<7;k++){
    float o = acc[k]+offb[k];
    o = (o-bnm[k])*rsqrtf(bnv[k]+EPSF)*bng[k]+bnb[k];
    y[k]=tanhf(o);
  }
  float d0=y[0]+y[1]+y[2], d1=y[1]+y[2], d2=y[2];
  float d4=y[4], d5=y[4]+y[5], d6=y[4]+y[5]+y[6];
  int h=h0+lr, w=w0+lc;
  size_t base = ((size_t)b*7)*HWS + (size_t)h*WW + w;
  dy[base        ] = d0; dy[base+  (size_t)HWS] = d1; dy[base+2*(size_t)HWS] = d2;
  dy[base+3*(size_t)HWS] = 0.f;
  dy[base+4*(size_t)HWS] = d4; dy[base+5*(size_t)HWS] = d5; dy[base+6*(size_t)HWS] = d6;
}

// ---------------- K6: deformable sample + implicit GEMM via f32 WMMA -> y (+GN partials) ----------------
__global__ void __launch_bounds__(256) kdsconv(
    const float* __restrict__ outb, const float* __restrict__ dy,
    const float* __restrict__ dsw, const float* __restrict__ dsb,
    float* __restrict__ yout, float* __restrict__ gnsum, float* __restrict__ gnsq){
  __shared__ float slab[32*8*41];   // [ci half (32)][row 8][col 41] of `out`
  __shared__ float At[32*34];       // A tile: [m (32)][ci (32)], padded stride 34
  __shared__ float Bt[64*34];       // B^T:   [co (64)][ci (32)], padded stride 34
  __shared__ float wrs[7*32];
  __shared__ int   i0s[7*32], i1s[7*32], ccs[7*32];
  __shared__ float gsum[16], gsq[16];
  int t = threadIdx.x;
  int w0 = blockIdx.x*32, h = blockIdx.y, b = blockIdx.z;

  // per-(k,m) vertical-lerp parameters: cols are integer -> pure row interpolation
  if(t < 224){
    int k = t>>5, m = t&31;
    int w = w0+m;
    float dv = dy[(((size_t)b*7+k)*HH_+h)*WW+w];
    float rv = fminf(fmaxf((float)h + dv, 0.f), 127.f);
    float r0f = floorf(rv);
    int r0 = (int)r0f;
    int r1 = mini(r0+1, 127);
    wrs[t] = rv - r0f;
    int base = h-3;
    i0s[t] = mini(maxi(r0-base,0),7);
    i1s[t] = mini(maxi(r1-base,0),7);
    int cg = mini(maxi(w + k - 3, 0), 127);
    ccs[t] = mini(maxi(cg - (w0-3), 0), 37);
  }
  if(t < 16){ gsum[t]=0.f; gsq[t]=0.f; }

  int lane = t & 31;
  int wv = t >> 5;
  int m0 = (wv >> 2) * 16;       // 2 M-tiles
  int n0 = (wv & 3) * 16;        // 4 N-tiles
  int lh = lane & 15;
  int hi = (lane >> 4) & 1;
  v8f acc = {0.f,0.f,0.f,0.f,0.f,0.f,0.f,0.f};

  for(int half=0; half<2; half++){
    int cb = half*32;
    __syncthreads();
    // stage 8x38 slab of `out` for 32 channels (handles all clamped gathers)
    for(int e=t; e<32*8*38; e+=256){
      int ci = e/304; int rem = e - ci*304; int r = rem/38; int c = rem - r*38;
      int gh = mini(maxi(h-3+r,0),127);
      int gw = mini(maxi(w0-3+c,0),127);
      slab[ci*328 + r*41 + c] = outb[(((size_t)b*CC + cb+ci)*HH_+gh)*WW+gw];
    }
    __syncthreads();
    for(int k=0;k<KS_;k++){
      { // build A tile (deform values) from slab via 1-D lerp
        int m = t & 31;
        int c4 = (t>>5)*4;
        float wr = wrs[k*32+m];
        int o0 = i0s[k*32+m]*41 + ccs[k*32+m];
        int o1 = i1s[k*32+m]*41 + ccs[k*32+m];
        #pragma unroll
        for(int e=0;e<4;e++){
          int ci = c4+e;
          float v0 = slab[ci*328+o0], v1 = slab[ci*328+o1];
          At[m*34+ci] = v0 + wr*(v1-v0);
        }
      }
      // load B^T slice: Bt[co][ci] = ds_w[co][cb+ci][k]
      for(int e=t; e<2048; e+=256){
        int co = e>>5, ci = e&31;
        Bt[co*34+ci] = dsw[((size_t)(co*CC)+(cb+ci))*KS_ + k];
      }
      __syncthreads();
      #pragma unroll
      for(int kc=0;kc<32;kc+=4){
        const float* ap = &At[(m0+lh)*34 + kc + hi*2];
        const float* bp = &Bt[(n0+lh)*34 + kc + hi*2];
        v2f af = {ap[0], ap[1]};
        v2f bf = {bp[0], bp[1]};
        acc = __builtin_amdgcn_wmma_f32_16x16x4_f32(
            false, af, false, bf, (short)0, acc, false, false);
      }
      __syncthreads();
    }
  }
  // epilogue: bias, store y, GroupNorm partial sums
  int co = n0 + lh;
  float bias = dsb[co];
  int grp = co >> 2;
  float ls=0.f, lq=0.f;
  #pragma unroll
  for(int i=0;i<8;i++){
    float v = acc[i] + bias;
    int m = m0 + hi*8 + i;
    yout[(((size_t)b*CC+co)*HH_+h)*WW + (w0+m)] = v;
    ls += v; lq += v*v;
  }
  atomicAdd(&gsum[grp], ls);
  atomicAdd(&gsq[grp], lq);
  __syncthreads();
  if(t<16){
    atomicAdd(&gnsum[b*16+t], gsum[t]);
    atomicAdd(&gnsq [b*16+t], gsq[t]);
  }
}

// ---------------- K7: GroupNorm finalize + ReLU (in place on d_out) ----------------
__global__ void kgn(float* __restrict__ y, const float* __restrict__ gnsum,
                    const float* __restrict__ gnsq, const float* __restrict__ gg,
                    const float* __restrict__ gb){
  size_t idx = (size_t)blockIdx.x*256 + threadIdx.x;
  int c = (int)((idx>>14) & 63);
  int b = (int)(idx>>20);
  int grp = c>>2;
  const float cnt = 4.f*(float)HWS;
  float s = gnsum[b*16+grp], q = gnsq[b*16+grp];
  float mu = s/cnt;
  float var = q/cnt - mu*mu;
  float v = (y[idx]-mu)*rsqrtf(var+EPSF)*gg[c] + gb[c];
  y[idx] = fmaxf(v, 0.f);
}

extern "C" void kernel_launch(void* const* d_in, const int* in_sizes, int n_in,
                              void* d_out, int out_size, void* d_ws, size_t ws_size,
                              hipStream_t stream){
  const float* x      = (const float*)d_in[0];
  const float* wgt    = (const float*)d_in[1];
  const float* ratio  = (const float*)d_in[2];
  const float* o1_fc  = (const float*)d_in[3];
  const float* o1_bg  = (const float*)d_in[4];
  const float* o1_bb  = (const float*)d_in[5];
  const float* o1_bm  = (const float*)d_in[6];
  const float* o1_bv  = (const float*)d_in[7];
  const float* o1_chw = (const float*)d_in[8];
  const float* o1_chb = (const float*)d_in[9];
  const float* o1_flw = (const float*)d_in[10];
  const float* o1_flb = (const float*)d_in[11];
  const float* o2_fc  = (const float*)d_in[12];
  const float* o2_bg  = (const float*)d_in[13];
  const float* o2_bb  = (const float*)d_in[14];
  const float* o2_bm  = (const float*)d_in[15];
  const float* o2_bv  = (const float*)d_in[16];
  const float* o2_chw = (const float*)d_in[17];
  const float* o2_chb = (const float*)d_in[18];
  const float* o2_flw = (const float*)d_in[19];
  const float* o2_flb = (const float*)d_in[20];
  const float* o2_spw = (const float*)d_in[21];
  const float* o2_spb = (const float*)d_in[22];
  const float* off_w  = (const float*)d_in[23];
  const float* off_b  = (const float*)d_in[24];
  const float* obn_g  = (const float*)d_in[25];
  const float* obn_b  = (const float*)d_in[26];
  const float* obn_m  = (const float*)d_in[27];
  const float* obn_v  = (const float*)d_in[28];
  const float* ds_w   = (const float*)d_in[29];
  const float* ds_b   = (const float*)d_in[30];
  const float* gn_g   = (const float*)d_in[31];
  const float* gn_b   = (const float*)d_in[32];

  float* ws   = (float*)d_ws;
  float* gbuf = ws;                 // 256
  float* c1   = ws + 256;           // 256
  float* f1   = ws + 512;
  float* c2   = ws + 768;
  float* f2   = ws + 1024;
  float* sp2  = ws + 1280;          // 64 (4x16 padded)
  float* weff = ws + 1344;          // 2304
  float* gnsm = ws + 3648;          // 64
  float* gnsq = ws + 3712;          // 64
  float* outb = ws + 4096;          // 4,194,304
  float* dybf = ws + 4096 + 4194304; // 458,752

  kpool<<<BB*CC, 256, 0, stream>>>(x, gbuf);
  katt<<<1, 256, 0, stream>>>(gbuf,
      o1_fc,o1_bg,o1_bb,o1_bm,o1_bv,o1_chw,o1_chb,o1_flw,o1_flb,
      o2_fc,o2_bg,o2_bb,o2_bm,o2_bv,o2_chw,o2_chb,o2_flw,o2_flb,
      o2_spw,o2_spb, c1,f1,c2,f2,sp2, gnsm);
  kweff<<<1, 256, 0, stream>>>(wgt, ratio, c1,f1,c2,f2, sp2, weff);
  kout<<<BB*CC, 256, 0, stream>>>(x, weff, outb);
  koff<<<dim3(8,8,BB), 256, 0, stream>>>(outb, off_w, off_b, obn_g,obn_b,obn_m,obn_v, dybf);
  kdsconv<<<dim3(4,HH_,BB), 256, 0, stream>>>(outb, dybf, ds_w, ds_b,
                                              (float*)d_out, gnsm, gnsq);
  kgn<<<(BB*CC*HWS)/256, 256, 0, stream>>>((float*)d_out, gnsm, gnsq, gn_g, gn_b);
}